// Integrator_24275155157296
// MI455X (gfx1250) — compile-verified
//
#include <hip/hip_runtime.h>
#include <cstdint>

#define KS   96
#define KM1  95
#define RPW  4            // rays per wave (pipeline depth)
#define WPB  8            // waves per block
#define BUFB 2048u        // bytes per staging buffer (1920 used, 16B aligned)

// ---- CDNA5 async global->LDS copy (ASYNCcnt-tracked) ---------------------
__device__ __forceinline__ void async_ld16(unsigned lds, unsigned long long ga) {
  asm volatile("global_load_async_to_lds_b128 %0, %1, off"
               :: "v"(lds), "v"(ga) : "memory");
}

// Stage one ray (rgb 1152B | sigma 384B | depth 384B) into LDS buffer.
// 5 async B128 ops per ray. Overshoot lanes: LDS addr out of range (write
// dropped per ISA), global addr clamped to a safe in-tensor address.
__device__ __forceinline__ void prefetch_ray(unsigned lbase,
                                             const float* __restrict__ rgbs,
                                             const float* __restrict__ sigmas,
                                             const float* __restrict__ depths,
                                             long ray, int lane) {
  unsigned long long rg = (unsigned long long)(uintptr_t)(rgbs   + (size_t)ray * (KS * 3));
  unsigned long long sg = (unsigned long long)(uintptr_t)(sigmas + (size_t)ray * KS);
  unsigned long long dg = (unsigned long long)(uintptr_t)(depths + (size_t)ray * KS);
  unsigned lo = (unsigned)lane * 16u;
  // rgb: 1152 bytes at buffer offset 0 (72 lanes worth -> 2 full + 8 lanes)
  async_ld16(lbase + lo,         rg + lo);
  async_ld16(lbase + 512u + lo,  rg + 512u + lo);
  {
    unsigned o2 = 1024u + lo;
    bool v = o2 < 1152u;
    async_ld16(v ? (lbase + o2) : 0x7ff00000u, v ? (rg + o2) : rg);
  }
  { // sigma: 384 bytes at offset 1152 (24 lanes)
    bool v = lo < 384u;
    async_ld16(v ? (lbase + 1152u + lo) : 0x7ff00000u, v ? (sg + lo) : sg);
  }
  { // depth: 384 bytes at offset 1536 (24 lanes)
    bool v = lo < 384u;
    async_ld16(v ? (lbase + 1536u + lo) : 0x7ff00000u, v ? (dg + lo) : dg);
  }
}

__device__ __forceinline__ float softplus_f(float x) {
  return fmaxf(x, 0.f) + log1pf(expf(-fabsf(x)));
}

__global__ __launch_bounds__(WPB * 32)
void nerf_composite(const float* __restrict__ rgbs, const float* __restrict__ sigmas,
                    const float* __restrict__ depths, float* __restrict__ out_rgb,
                    float* __restrict__ out_depth, float* __restrict__ out_w,
                    unsigned* __restrict__ ws, int num_rays) {
  __shared__ __align__(16) unsigned char smem[WPB * 2 * BUFB];
  __shared__ float smin[WPB], smax[WPB];

  const int lane = (int)(threadIdx.x & 31);
  const int wave = (int)(threadIdx.x >> 5);
  const long ray0 = ((long)blockIdx.x * WPB + wave) * RPW;
  const unsigned lbase = (unsigned)(uintptr_t)&smem[0] + (unsigned)wave * (2u * BUFB);

  float vmin = __int_as_float(0x7f800000);  // +inf  (lane 0 tracks min first-midpoint)
  float vmax = 0.0f;                        // all depths > 0 (lane 31 tracks max last-midpoint)

  long pr0 = ray0 < (long)num_rays ? ray0 : (long)num_rays - 1;
  prefetch_ray(lbase, rgbs, sigmas, depths, pr0, lane);

  for (int i = 0; i < RPW; ++i) {
    long ray = ray0 + i;
    if (i + 1 < RPW) {
      long nray = (ray + 1) < (long)num_rays ? (ray + 1) : (long)num_rays - 1;
      prefetch_ray(lbase + (unsigned)((i + 1) & 1) * BUFB, rgbs, sigmas, depths, nray, lane);
      asm volatile("s_wait_asynccnt 5" ::: "memory");   // current ray's 5 ops done
    } else {
      asm volatile("s_wait_asynccnt 0" ::: "memory");
    }
    if (ray >= (long)num_rays) continue;

    const float* b = (const float*)(smem + (size_t)wave * (2u * BUFB) + (size_t)(i & 1) * BUFB);
    const int k0 = lane * 3;

    // lane L owns samples 3L,3L+1,3L+2 ; sample 3L+3 via shuffle from lane L+1
    float r0 = b[9 * lane + 0], g0 = b[9 * lane + 1], c0 = b[9 * lane + 2];
    float r1 = b[9 * lane + 3], g1 = b[9 * lane + 4], c1 = b[9 * lane + 5];
    float r2 = b[9 * lane + 6], g2 = b[9 * lane + 7], c2 = b[9 * lane + 8];
    float s0 = b[288 + k0], s1 = b[288 + k0 + 1], s2 = b[288 + k0 + 2];
    float d0 = b[384 + k0], d1 = b[384 + k0 + 1], d2 = b[384 + k0 + 2];
    float rn = __shfl_down(r0, 1, 32), gn = __shfl_down(g0, 1, 32), cn = __shfl_down(c0, 1, 32);
    float sn = __shfl_down(s0, 1, 32), dn = __shfl_down(d0, 1, 32);
    const bool has2 = (lane < 31);  // 95 outputs total: lane31 emits only 2

    float sp0 = softplus_f(0.5f * (s0 + s1) - 1.0f);
    float sp1 = softplus_f(0.5f * (s1 + s2) - 1.0f);
    float sp2 = softplus_f(0.5f * (s2 + sn) - 1.0f);
    float a0 = -expm1f(-(d1 - d0) * sp0);
    float a1 = -expm1f(-(d2 - d1) * sp1);
    float a2 = has2 ? -expm1f(-(dn - d2) * sp2) : 0.f;
    float t0 = 1.f - a0 + 1e-10f;
    float t1 = 1.f - a1 + 1e-10f;
    float t2 = has2 ? (1.f - a2 + 1e-10f) : 1.f;

    // exclusive multiplicative wave scan -> transmittance
    float acc = t0 * t1 * t2;
#pragma unroll
    for (int o = 1; o < 32; o <<= 1) {
      float v = __shfl_up(acc, o, 32);
      if (lane >= o) acc *= v;
    }
    float excl = __shfl_up(acc, 1, 32);
    if (lane == 0) excl = 1.f;

    float w0 = a0 * excl;
    float w1 = a1 * excl * t0;
    float w2 = a2 * excl * t0 * t1;

    float dm0 = 0.5f * (d0 + d1), dm1 = 0.5f * (d1 + d2), dm2 = 0.5f * (d2 + dn);
    float wsum = w0 + w1 + w2;
    float cr = w0 * 0.5f * (r0 + r1) + w1 * 0.5f * (r1 + r2) + w2 * 0.5f * (r2 + rn);
    float cg = w0 * 0.5f * (g0 + g1) + w1 * 0.5f * (g1 + g2) + w2 * 0.5f * (g2 + gn);
    float cb = w0 * 0.5f * (c0 + c1) + w1 * 0.5f * (c1 + c2) + w2 * 0.5f * (c2 + cn);
    float cd = w0 * dm0 + w1 * dm1 + w2 * dm2;
#pragma unroll
    for (int o = 16; o >= 1; o >>= 1) {
      wsum += __shfl_xor(wsum, o, 32);
      cr   += __shfl_xor(cr, o, 32);
      cg   += __shfl_xor(cg, o, 32);
      cb   += __shfl_xor(cb, o, 32);
      cd   += __shfl_xor(cd, o, 32);
    }

    float* wp = out_w + (size_t)ray * KM1 + k0;
    wp[0] = w0;
    wp[1] = w1;
    if (has2) wp[2] = w2;

    if (lane == 0) {
      out_rgb[(size_t)ray * 3 + 0] = cr * 2.f - 1.f;
      out_rgb[(size_t)ray * 3 + 1] = cg * 2.f - 1.f;
      out_rgb[(size_t)ray * 3 + 2] = cb * 2.f - 1.f;
      float dd = cd / wsum;
      if (!(dd == dd)) dd = __int_as_float(0x7f800000);  // nan -> +inf (clamped later)
      out_depth[ray] = dd;
      vmin = fminf(vmin, dm0);   // depths sorted: first midpoint = ray min
    }
    if (lane == 31) vmax = fmaxf(vmax, dm1);  // (94,95) midpoint = ray max
  }

  if (lane == 0)  smin[wave] = vmin;
  if (lane == 31) smax[wave] = vmax;
  __syncthreads();
  if (threadIdx.x == 0) {
    float m = smin[0], M = smax[0];
#pragma unroll
    for (int wv = 1; wv < WPB; ++wv) { m = fminf(m, smin[wv]); M = fmaxf(M, smax[wv]); }
    atomicMin(ws + 0, __float_as_uint(m));   // positive floats: uint order == float order
    atomicMax(ws + 1, __float_as_uint(M));
  }
}

__global__ void init_ws(unsigned* ws) {
  ws[0] = 0x7f800000u;  // +inf
  ws[1] = 0u;           // 0.0f
}

__global__ void clamp_depth(float* __restrict__ out_depth, const unsigned* __restrict__ ws, int n) {
  int i = blockIdx.x * blockDim.x + threadIdx.x;
  if (i < n) {
    float lo = __uint_as_float(ws[0]);
    float hi = __uint_as_float(ws[1]);
    float v = out_depth[i];
    out_depth[i] = fminf(fmaxf(v, lo), hi);
  }
}

extern "C" void kernel_launch(void* const* d_in, const int* in_sizes, int n_in,
                              void* d_out, int out_size, void* d_ws, size_t ws_size,
                              hipStream_t stream) {
  const float* rgbs   = (const float*)d_in[0];
  const float* sigmas = (const float*)d_in[1];
  const float* depths = (const float*)d_in[2];

  int num_rays = in_sizes[1] / KS;  // sigmas: N*R*K -> N*R rays

  float* out       = (float*)d_out;
  float* out_rgb   = out;                              // [rays,3]
  float* out_depth = out + (size_t)num_rays * 3;       // [rays,1]
  float* out_w     = out + (size_t)num_rays * 4;       // [rays,95]
  unsigned* ws     = (unsigned*)d_ws;

  init_ws<<<1, 1, 0, stream>>>(ws);

  int rays_per_block = RPW * WPB;  // 32
  int grid = (num_rays + rays_per_block - 1) / rays_per_block;
  nerf_composite<<<grid, WPB * 32, 0, stream>>>(rgbs, sigmas, depths,
                                                out_rgb, out_depth, out_w, ws, num_rays);

  clamp_depth<<<(num_rays + 255) / 256, 256, 0, stream>>>(out_depth, ws, num_rays);
}